// LatentSpace_71554155151615
// MI455X (gfx1250) — compile-verified
//
#include <hip/hip_runtime.h>
#include <hip/hip_bf16.h>
#include <math.h>

typedef __attribute__((ext_vector_type(2))) float v2f;
typedef __attribute__((ext_vector_type(8))) float v8f;

#define NCAT   136
#define NPAD   144
#define DDIM   16
#define NSAMP  20480
#define LOG2PI_C 1.8378770664093453f

// ---- workspace layout (floats) ----
#define WS_ALPHA   0      // alpha[16][144]
#define WS_MUB     2304   // muB  [16][144]
#define WS_INVSIG  4608   // invsig[144]
#define WS_RECIP   4752   // recip [144]
#define WS_SQC     4896   // sqrt(clip(invsig)) [144]
#define WS_LOGTERM 5040   // 0.5*(LOG2PI - log(clip(invsig))) [144]
#define WS_DOTAM   5184   // alpha . muB [144]
#define WS_MUBSQ   5328   // |muB|^2 [144]
#define WS_LOGPI   5472   // log(clip(softmax(pi))) [144] (pads = -1e30)
#define WS_PART    5616   // block partials [160]

__global__ __launch_bounds__(256) void prep_kernel(const float* __restrict__ pi,
                                                   const float* __restrict__ mu,
                                                   float* __restrict__ ws) {
  __shared__ float sm[NCAT];
  __shared__ float smax, slse;
  int c = threadIdx.x;
  if (c < NCAT) sm[c] = pi[c];
  __syncthreads();
  if (c == 0) {
    float m = -3.0e38f;
    for (int i = 0; i < NCAT; ++i) m = fmaxf(m, sm[i]);
    float s = 0.f;
    for (int i = 0; i < NCAT; ++i) s += __expf(sm[i] - m);
    smax = m; slse = __logf(s);
  }
  __syncthreads();
  if (c < NPAD) {
    float lpi, invsig = 0.f, dam = 0.f, mb2 = 0.f;
    if (c < NCAT) {
      // row-major upper-triangle enumeration: (a,b), a<=b
      int a = 0, cum = 0;
      while (c >= cum + (DDIM - a)) { cum += DDIM - a; ++a; }
      int b = a + (c - cum);
      for (int d = 0; d < DDIM; ++d) {
        float mB = mu[d * DDIM + b];
        float mA = mu[d * DDIM + a];
        float al = mB - mA;
        ws[WS_ALPHA + d * NPAD + c] = al;
        ws[WS_MUB   + d * NPAD + c] = mB;
        invsig += al * al;
        dam    += al * mB;
        mb2    += mB * mB;
      }
      lpi = fmaxf(sm[c] - smax - slse, -13.815510558f /* log(1e-6) */);
    } else {
      for (int d = 0; d < DDIM; ++d) {
        ws[WS_ALPHA + d * NPAD + c] = 0.f;
        ws[WS_MUB   + d * NPAD + c] = 0.f;
      }
      lpi = -1.0e30f; // padding category vanishes in logsumexp
    }
    ws[WS_INVSIG + c] = invsig;
    ws[WS_RECIP  + c] = (invsig == 0.f) ? 0.f : 1.f / invsig;
    float cl = fminf(fmaxf(invsig, 1e-12f), 1e30f);
    ws[WS_SQC     + c] = sqrtf(cl);
    ws[WS_LOGTERM + c] = 0.5f * (LOG2PI_C - __logf(cl));
    ws[WS_DOTAM   + c] = dam;
    ws[WS_MUBSQ   + c] = mb2;
    ws[WS_LOGPI   + c] = lpi;
  }
}

// 8 waves/block, each wave owns a 16-sample tile: grid = 20480/16/8 = 160 blocks.
__global__ __launch_bounds__(256) void latent_main_kernel(const float* __restrict__ z,
                                                          const float* __restrict__ ws,
                                                          float* __restrict__ partials) {
  const float* alpha   = ws + WS_ALPHA;
  const float* muB     = ws + WS_MUB;
  const float* invsig  = ws + WS_INVSIG;
  const float* recip   = ws + WS_RECIP;
  const float* sqc     = ws + WS_SQC;
  const float* logterm = ws + WS_LOGTERM;
  const float* dot_am  = ws + WS_DOTAM;
  const float* muBsq   = ws + WS_MUBSQ;
  const float* logpi   = ws + WS_LOGPI;

  const int lane = threadIdx.x & 31;
  const int wave = threadIdx.x >> 5;
  const int half = lane >> 4;     // 0: lanes 0-15, 1: lanes 16-31
  const int l16  = lane & 15;
  const int base = (blockIdx.x * 8 + wave) * 16;   // first sample of tile

  // ---- A fragments: Z tile, 16x16 f32 as 4 K-steps of 16x4 ----
  // layout: lane holds m=l16; vgpr j of step kk holds k = 4*kk + j + 2*half
  v2f afrag[4];
#pragma unroll
  for (int kk = 0; kk < 4; ++kk)
#pragma unroll
    for (int j = 0; j < 2; ++j)
      afrag[kk][j] = z[(base + l16) * DDIM + 4 * kk + j + 2 * half];

  // per-sample |z|^2: each lane holds 8 of 16 dims for sample l16
  float part = 0.f;
#pragma unroll
  for (int kk = 0; kk < 4; ++kk)
#pragma unroll
    for (int j = 0; j < 2; ++j)
      part += afrag[kk][j] * afrag[kk][j];
  float zsq_all = part + __shfl_xor(part, 16, 32);   // lane holds zsq[sample l16]

  // hoisted loop-invariant broadcast: zq8[r] = |z|^2 of sample m = r + 8*half
  float zq8[8];
#pragma unroll
  for (int r = 0; r < 8; ++r)
    zq8[r] = __shfl(zsq_all, r + 8 * half, 32);

  // streaming logsumexp state, 8 rows per lane (samples m = r + 8*half)
  float rmax[8], rsum[8];
#pragma unroll
  for (int r = 0; r < 8; ++r) { rmax[r] = -3.0e38f; rsum[r] = 0.f; }

  for (int ct = 0; ct < NPAD / 16; ++ct) {          // 9 category tiles
    const int cbase = ct * 16;
    // B fragments: lane holds n = l16; vgpr j of step kk holds k = 4*kk + j + 2*half
    v2f bA[4], bM[4];
#pragma unroll
    for (int kk = 0; kk < 4; ++kk)
#pragma unroll
      for (int j = 0; j < 2; ++j) {
        int d = 4 * kk + j + 2 * half;
        int c = cbase + l16;
        bA[kk][j] = alpha[d * NPAD + c];
        bM[kk][j] = muB[d * NPAD + c];
      }

    v8f acc1 = {};   // Z . Alpha
    v8f acc2 = {};   // Z . MuB
#pragma unroll
    for (int kk = 0; kk < 4; ++kk) {
      acc1 = __builtin_amdgcn_wmma_f32_16x16x4_f32(false, afrag[kk], false, bA[kk],
                                                   (short)0, acc1, false, false);
      acc2 = __builtin_amdgcn_wmma_f32_16x16x4_f32(false, afrag[kk], false, bM[kk],
                                                   (short)0, acc2, false, false);
    }

    const int   c   = cbase + l16;
    const float isg = invsig[c];
    const float rcp = recip[c];
    const float sq  = sqc[c];
    const float lt  = logterm[c];
    const float dam = dot_am[c];
    const float mb2 = muBsq[c];
    const float lpi = logpi[c];

#pragma unroll
    for (int r = 0; r < 8; ++r) {
      // C/D layout: vgpr r -> sample m = r + 8*half, category n = l16
      float dotA = acc1[r];
      float dotM = acc2[r];
      float s_ab = dotA - dam;                 // alpha . (z - muB)
      float nu   = -s_ab * rcp;
      float bsq  = zq8[r] - 2.f * dotM + mb2;  // |z - muB|^2
      float t    = -bsq + nu * nu * isg;
      float le0  = lt + 0.5f * t;
      float e1   = (1.f - nu) * sq;
      float e2   = -nu * sq;
      // Phi(x) = 0.5*erfc(-x/sqrt(2))
      float diff = 0.5f * (erfcf(-e1 * 0.70710678118654752f)
                         - erfcf(-e2 * 0.70710678118654752f));
      diff = fminf(fmaxf(diff, 1e-6f), 1e30f);
      float lpzc = le0 + __logf(diff);
      float val  = -8.0f * LOG2PI_C + lpi
                 + ((isg == 0.f) ? (-0.5f * bsq) : lpzc);
      // branchless streaming logsumexp update
      float nm = fmaxf(rmax[r], val);
      rsum[r] = rsum[r] * __expf(rmax[r] - nm) + __expf(val - nm);
      rmax[r] = nm;
    }
  }

  // merge (max,sum) across the 16 lanes of each half (categories)
#pragma unroll
  for (int mask = 1; mask < 16; mask <<= 1) {
#pragma unroll
    for (int r = 0; r < 8; ++r) {
      float om = __shfl_xor(rmax[r], mask, 32);
      float os = __shfl_xor(rsum[r], mask, 32);
      float nm = fmaxf(rmax[r], om);
      rsum[r]  = rsum[r] * __expf(rmax[r] - nm) + os * __expf(om - nm);
      rmax[r]  = nm;
    }
  }

  // lanes 0 and 16 each own 8 distinct samples; sum their log p(z)
  float local = 0.f;
  if (l16 == 0) {
#pragma unroll
    for (int r = 0; r < 8; ++r) local += rmax[r] + __logf(rsum[r]);
  }
  local += __shfl_xor(local, 16, 32);   // lane 0 now has the wave's 16-sample sum

  __shared__ float sred[8];
  if (lane == 0) sred[wave] = local;
  __syncthreads();
  if (threadIdx.x == 0) {
    float s = 0.f;
    for (int w = 0; w < 8; ++w) s += sred[w];
    partials[blockIdx.x] = s;
  }
}

__global__ __launch_bounds__(256) void final_kernel(const float* __restrict__ partials,
                                                    float* __restrict__ out,
                                                    int nblocks, float invN) {
  __shared__ float s[256];
  float v = 0.f;
  for (int i = threadIdx.x; i < nblocks; i += 256) v += partials[i];
  s[threadIdx.x] = v;
  __syncthreads();
  for (int st = 128; st > 0; st >>= 1) {
    if (threadIdx.x < st) s[threadIdx.x] += s[threadIdx.x + st];
    __syncthreads();
  }
  if (threadIdx.x == 0) out[0] = s[0] * invN;
}

extern "C" void kernel_launch(void* const* d_in, const int* in_sizes, int n_in,
                              void* d_out, int out_size, void* d_ws, size_t ws_size,
                              hipStream_t stream) {
  const float* z  = (const float*)d_in[0];  // (2048,10,16) = 20480 x 16
  const float* pi = (const float*)d_in[1];  // 136
  const float* mu = (const float*)d_in[2];  // 16 x 16
  float* ws  = (float*)d_ws;
  float* out = (float*)d_out;

  prep_kernel<<<1, 256, 0, stream>>>(pi, mu, ws);
  latent_main_kernel<<<160, 256, 0, stream>>>(z, ws, ws + WS_PART);
  final_kernel<<<1, 256, 0, stream>>>(ws + WS_PART, out, 160, 1.0f / (float)NSAMP);
}